// Network_21010980012233
// MI455X (gfx1250) — compile-verified
//
#include <hip/hip_runtime.h>
#include <math.h>

#define BN_RS 0.99995000374968755f   // 1/sqrt(1 + 1e-4)
#define SX 34                        // LDS row stride for K=32 tiles
#define SE 66                        // LDS row stride for K=64 tiles
#define PK_WAVES 8                   // waves per block in point kernel
#define SP_WAVES 4                   // waves per block in superpoint kernels

typedef float v2f __attribute__((ext_vector_type(2)));
typedef float v8f __attribute__((ext_vector_type(8)));

// ---------------------------------------------------------------------------
// fp32 WMMA: D(16x16) += A(16x4) * B(4x16), accumulated over K in steps of 4.
// A layout: lane<16 holds row M=lane, K={k,k+1}; lane>=16 holds K={k+2,k+3}.
// B layout (mirror): VGPR0 = rows {k, k+2} striped over lane halves, VGPR1 = {k+1, k+3}.
// D layout: acc[r] @ lane l -> (M = r + 8*(l>>4), N = l&15).
// ---------------------------------------------------------------------------
__device__ __forceinline__ v8f wmma4(v2f a, v2f b, v8f c) {
  return __builtin_amdgcn_wmma_f32_16x16x4_f32(false, a, false, b, (short)0, c,
                                               false, false);
}

// D(16x16) = X(16xK, LDS, row stride sx) @ W^T, with W row-major [nW x K],
// producing output columns n0..n0+15 (columns >= nW are zero-padded).
__device__ __forceinline__ v8f tile_gemm(const float* Xs, int sx,
                                         const float* __restrict__ W,
                                         int K, int n0, int nW) {
  const int lane = threadIdx.x & 31;
  const int half = lane >> 4;
  const int idx  = lane & 15;
  const int n    = n0 + idx;
  v8f acc;
#pragma unroll
  for (int i = 0; i < 8; ++i) acc[i] = 0.0f;
  for (int k = 0; k < K; k += 4) {
    v2f a = *(const v2f*)(Xs + idx * sx + k + 2 * half);
    v2f b;
    if (n < nW) {
      b = *(const v2f*)(W + n * K + k + 2 * half);
    } else {
      b.x = 0.0f; b.y = 0.0f;
    }
    acc = wmma4(a, b, acc);
  }
  return acc;
}

// ordered-uint mapping so float max can use integer atomicMax
__device__ __forceinline__ unsigned f2o(float f) {
  unsigned u = __float_as_uint(f);
  return (u & 0x80000000u) ? ~u : (u | 0x80000000u);
}
__device__ __forceinline__ float o2f(unsigned u) {
  return (u & 0x80000000u) ? __uint_as_float(u & 0x7fffffffu)
                           : __uint_as_float(~u);
}

// ---------------------------------------------------------------------------
// init: zero accumulators, set segment-max to -inf (ordered encoding)
// ---------------------------------------------------------------------------
__global__ void k_init(float* emb_sum, float* cnt, float* segsum,
                       unsigned* segmax, float* res, int S) {
  int i = blockIdx.x * blockDim.x + threadIdx.x;
  if (i < S * 64) res[i] = 0.0f;
  if (i < S * 32) emb_sum[i] = 0.0f;
  if (i < S) {
    cnt[i] = 0.0f;
    segsum[i] = 0.0f;
    segmax[i] = f2o(-__builtin_huge_valf());
  }
}

// ---------------------------------------------------------------------------
// point kernel: gather voxel feats, scatter-add to superpoints, 32->32->20 MLP
// ---------------------------------------------------------------------------
__global__ void k_point(const float* __restrict__ feats,
                        const int* __restrict__ imap,
                        const int* __restrict__ spof,
                        const float* __restrict__ W1, const float* __restrict__ b1,
                        const float* __restrict__ g,  const float* __restrict__ be,
                        const float* __restrict__ W2, const float* __restrict__ b2,
                        float* __restrict__ sem_out, float* __restrict__ emb_sum,
                        float* __restrict__ cnt, int Np) {
  __shared__ __attribute__((aligned(16))) float Xs[PK_WAVES][16 * SX];
  __shared__ __attribute__((aligned(16))) float Hs[PK_WAVES][16 * SX];
  const int wave = threadIdx.x >> 5;
  const int lane = threadIdx.x & 31;
  const int half = lane >> 4;
  const int idx  = lane & 15;
  const int p0   = (blockIdx.x * PK_WAVES + wave) * 16;
  float* xs = Xs[wave];
  float* hs = Hs[wave];

  for (int m = 0; m < 16; ++m) {
    int p = p0 + m;
    float x = 0.0f;
    if (p < Np) {
      x = feats[imap[p] * 32 + lane];
      int sp = spof[p];
      atomicAdd(&emb_sum[sp * 32 + lane], x);
      if (lane == 0) atomicAdd(&cnt[sp], 1.0f);
    }
    xs[m * SX + lane] = x;
  }
  __syncthreads();

  // Linear(32->32) -> BN -> ReLU
  for (int nt = 0; nt < 2; ++nt) {
    v8f acc = tile_gemm(xs, SX, W1, 32, nt * 16, 32);
    int n = nt * 16 + idx;
    float sc = g[n] * BN_RS;
    float bi = b1[n] * sc + be[n];
#pragma unroll
    for (int r = 0; r < 8; ++r) {
      int m = r + 8 * half;
      hs[m * SX + n] = fmaxf(acc[r] * sc + bi, 0.0f);
    }
  }
  __syncthreads();

  // Linear(32->20)
  for (int nt = 0; nt < 2; ++nt) {
    v8f acc = tile_gemm(hs, SX, W2, 32, nt * 16, 20);
    int n = nt * 16 + idx;
    if (n < 20) {
#pragma unroll
      for (int r = 0; r < 8; ++r) {
        int m = r + 8 * half;
        int p = p0 + m;
        if (p < Np) sem_out[p * 20 + n] = acc[r] + b2[n];
      }
    }
  }
}

// 64 -> 64 (BN+ReLU) -> nout head, X already in LDS (es), hs is scratch LDS
__device__ void head64(const float* es, float* hs,
                       const float* __restrict__ W1, const float* __restrict__ b1,
                       const float* __restrict__ g,  const float* __restrict__ be,
                       const float* __restrict__ W2, const float* __restrict__ b2,
                       int nout, float* __restrict__ out, int ldo,
                       int s0, int S) {
  const int lane = threadIdx.x & 31;
  const int half = lane >> 4;
  const int idx  = lane & 15;
  for (int nt = 0; nt < 4; ++nt) {
    v8f acc = tile_gemm(es, SE, W1, 64, nt * 16, 64);
    int n = nt * 16 + idx;
    float sc = g[n] * BN_RS;
    float bi = b1[n] * sc + be[n];
#pragma unroll
    for (int r = 0; r < 8; ++r) {
      int m = r + 8 * half;
      hs[m * SE + n] = fmaxf(acc[r] * sc + bi, 0.0f);
    }
  }
  __syncthreads();
  int ntiles = (nout + 15) >> 4;
  for (int nt = 0; nt < ntiles; ++nt) {
    v8f acc = tile_gemm(hs, SE, W2, 64, nt * 16, nout);
    int n = nt * 16 + idx;
    if (n < nout) {
#pragma unroll
      for (int r = 0; r < 8; ++r) {
        int m = r + 8 * half;
        int s = s0 + m;
        if (s < S) out[s * ldo + n] = acc[r] + b2[n];
      }
    }
  }
  __syncthreads();
}

// ---------------------------------------------------------------------------
// superpoint kernel: emb -> ecc -> q/k/v + 4 small heads (all WMMA)
// ---------------------------------------------------------------------------
__global__ void k_sp(const float* __restrict__ emb_sum, const float* __restrict__ cnt,
                     const float* __restrict__ W_ecc,
                     const float* __restrict__ Wq, const float* __restrict__ Wk,
                     const float* __restrict__ Wv,
                     const float* sW1, const float* sb1, const float* sg,
                     const float* sbe, const float* sW2, const float* sb2,
                     const float* oW1, const float* ob1, const float* og,
                     const float* obe, const float* oW2, const float* ob2,
                     const float* cW1, const float* cb1, const float* cg,
                     const float* cbe, const float* cW2, const float* cb2,
                     const float* zW1, const float* zb1, const float* zg,
                     const float* zbe, const float* zW2, const float* zb2,
                     float* __restrict__ ecc, float* __restrict__ q,
                     float* __restrict__ kk, float* __restrict__ v,
                     float* sp_sem, float* sp_off, float* sp_occ, float* sp_size,
                     int S) {
  __shared__ __attribute__((aligned(16))) float Xs[SP_WAVES][16 * SX];
  __shared__ __attribute__((aligned(16))) float Es[SP_WAVES][16 * SE];
  __shared__ __attribute__((aligned(16))) float Hs[SP_WAVES][16 * SE];
  const int wave = threadIdx.x >> 5;
  const int lane = threadIdx.x & 31;
  const int half = lane >> 4;
  const int idx  = lane & 15;
  const int s0   = (blockIdx.x * SP_WAVES + wave) * 16;
  float* xs = Xs[wave];
  float* es = Es[wave];
  float* hs = Hs[wave];

  // emb = segment_sum / max(cnt,1)
  for (int m = 0; m < 16; ++m) {
    int s = s0 + m;
    float x = 0.0f;
    if (s < S) x = emb_sum[s * 32 + lane] / fmaxf(cnt[s], 1.0f);
    xs[m * SX + lane] = x;
  }
  __syncthreads();

  // ecc = emb @ W_ecc.T   (32 -> 64)
  for (int nt = 0; nt < 4; ++nt) {
    v8f acc = tile_gemm(xs, SX, W_ecc, 32, nt * 16, 64);
    int n = nt * 16 + idx;
#pragma unroll
    for (int r = 0; r < 8; ++r) {
      int m = r + 8 * half;
      es[m * SE + n] = acc[r];
      int s = s0 + m;
      if (s < S) ecc[s * 64 + n] = acc[r];
    }
  }
  __syncthreads();

  // q, k, v = ecc @ W.T   (64 -> 64 each)
  const float* Ws[3] = {Wq, Wk, Wv};
  float* Os[3] = {q, kk, v};
  for (int t = 0; t < 3; ++t) {
    for (int nt = 0; nt < 4; ++nt) {
      v8f acc = tile_gemm(es, SE, Ws[t], 64, nt * 16, 64);
      int n = nt * 16 + idx;
#pragma unroll
      for (int r = 0; r < 8; ++r) {
        int m = r + 8 * half;
        int s = s0 + m;
        if (s < S) Os[t][s * 64 + n] = acc[r];
      }
    }
  }
  __syncthreads();

  head64(es, hs, sW1, sb1, sg, sbe, sW2, sb2, 20, sp_sem, 20, s0, S);
  head64(es, hs, oW1, ob1, og, obe, oW2, ob2, 3,  sp_off, 3,  s0, S);
  head64(es, hs, cW1, cb1, cg, cbe, cW2, cb2, 1,  sp_occ, 1,  s0, S);
  head64(es, hs, zW1, zb1, zg, zbe, zW2, zb2, 1,  sp_size, 1, s0, S);
}

// ---------------------------------------------------------------------------
// edge pass 1: aff = (q[u].k[v])/8 * posMLP(xyz[u]-xyz[v]); segment max
// ---------------------------------------------------------------------------
__global__ void k_edge_aff(const int* __restrict__ eu, const int* __restrict__ ev,
                           const float* __restrict__ xyz,
                           const float* __restrict__ pW1, const float* __restrict__ pb1,
                           const float* __restrict__ pW2, const float* __restrict__ pb2,
                           const float* __restrict__ q, const float* __restrict__ k,
                           float* __restrict__ aff, unsigned* __restrict__ segmax,
                           int E) {
  int e = blockIdx.x * blockDim.x + threadIdx.x;
  if (e >= E) return;
  int u = eu[e], w = ev[e];
  float d0 = xyz[u * 3 + 0] - xyz[w * 3 + 0];
  float d1 = xyz[u * 3 + 1] - xyz[w * 3 + 1];
  float d2 = xyz[u * 3 + 2] - xyz[w * 3 + 2];
  float pos = pb2[0];
#pragma unroll
  for (int j = 0; j < 16; ++j) {
    float h = fmaxf(d0 * pW1[j * 3 + 0] + d1 * pW1[j * 3 + 1] +
                    d2 * pW1[j * 3 + 2] + pb1[j], 0.0f);
    pos += h * pW2[j];
  }
  float dot = 0.0f;
  const float4* qr = (const float4*)(q + (size_t)u * 64);
  const float4* kr = (const float4*)(k + (size_t)w * 64);
#pragma unroll
  for (int j = 0; j < 16; ++j) {
    float4 a = qr[j], b = kr[j];
    dot += a.x * b.x + a.y * b.y + a.z * b.z + a.w * b.w;
  }
  float af = dot * 0.125f * pos;
  aff[e] = af;
  atomicMax(&segmax[u], f2o(af));
}

// edge pass 2: ex = exp(aff - mx[u]); segment sum
__global__ void k_edge_exp(const int* __restrict__ eu, float* __restrict__ aff,
                           const unsigned* __restrict__ segmax,
                           float* __restrict__ segsum, int E) {
  int e = blockIdx.x * blockDim.x + threadIdx.x;
  if (e >= E) return;
  int u = eu[e];
  float ex = expf(aff[e] - o2f(segmax[u]));
  aff[e] = ex;
  atomicAdd(&segsum[u], ex);
}

// edge pass 3 (one wave per edge): soft = ex/tot; res[u] += soft * v[v]
__global__ void k_edge_soft(const int* __restrict__ eu, const int* __restrict__ ev,
                            const float* __restrict__ ex, const float* __restrict__ segsum,
                            const float* __restrict__ v, float* __restrict__ soft_out,
                            float* __restrict__ res, int E) {
  int e = (blockIdx.x * blockDim.x + threadIdx.x) >> 5;
  int lane = threadIdx.x & 31;
  if (e >= E) return;
  int u = eu[e], w = ev[e];
  float s = ex[e] / segsum[u];
  if (lane == 0) soft_out[e] = s;
  float a = v[(size_t)w * 64 + lane];
  float b = v[(size_t)w * 64 + 32 + lane];
  atomicAdd(&res[(size_t)u * 64 + lane], s * a);
  atomicAdd(&res[(size_t)u * 64 + 32 + lane], s * b);
}

// ---------------------------------------------------------------------------
// final: sp_feat = ecc + res -> 64->64->7 head
// ---------------------------------------------------------------------------
__global__ void k_final(const float* __restrict__ ecc, const float* __restrict__ res,
                        const float* fW1, const float* fb1, const float* fg,
                        const float* fbe, const float* fW2, const float* fb2,
                        float* __restrict__ sp_disc, int S) {
  __shared__ __attribute__((aligned(16))) float Es[SP_WAVES][16 * SE];
  __shared__ __attribute__((aligned(16))) float Hs[SP_WAVES][16 * SE];
  const int wave = threadIdx.x >> 5;
  const int lane = threadIdx.x & 31;
  const int s0   = (blockIdx.x * SP_WAVES + wave) * 16;
  float* es = Es[wave];
  float* hs = Hs[wave];
  for (int m = 0; m < 16; ++m) {
    int s = s0 + m;
    float x0 = 0.0f, x1 = 0.0f;
    if (s < S) {
      x0 = ecc[s * 64 + lane]      + res[s * 64 + lane];
      x1 = ecc[s * 64 + 32 + lane] + res[s * 64 + 32 + lane];
    }
    es[m * SE + lane]      = x0;
    es[m * SE + 32 + lane] = x1;
  }
  __syncthreads();
  head64(es, hs, fW1, fb1, fg, fbe, fW2, fb2, 7, sp_disc, 7, s0, S);
}

// ---------------------------------------------------------------------------
#define F(i) ((const float*)d_in[i])
#define HEADP(i) F(i), F(i + 1), F(i + 2), F(i + 3), F(i + 4), F(i + 5)

extern "C" void kernel_launch(void* const* d_in, const int* in_sizes, int n_in,
                              void* d_out, int out_size, void* d_ws, size_t ws_size,
                              hipStream_t stream) {
  const float* feats = F(0);
  const int* imap = (const int*)d_in[1];
  const int* spof = (const int*)d_in[2];
  const float* xyz = F(3);
  const int* eu = (const int*)d_in[4];
  const int* ev = (const int*)d_in[5];
  const float* W_ecc = F(6);

  const int Np = in_sizes[1];
  const int S  = in_sizes[3] / 3;
  const int E  = in_sizes[4];

  // outputs, concatenated in reference return order
  float* out = (float*)d_out;
  float* o_sem    = out;
  float* o_spsem  = o_sem + (size_t)Np * 20;
  float* o_spoff  = o_spsem + (size_t)S * 20;
  float* o_spocc  = o_spoff + (size_t)S * 3;
  float* o_spsize = o_spocc + S;
  float* o_soft   = o_spsize + S;
  float* o_spdisc = o_soft + E;

  // workspace carve
  float* ws = (float*)d_ws;
  float* emb_sum = ws; ws += (size_t)S * 32;
  float* cnt     = ws; ws += S;
  float* ecc     = ws; ws += (size_t)S * 64;
  float* q       = ws; ws += (size_t)S * 64;
  float* k       = ws; ws += (size_t)S * 64;
  float* v       = ws; ws += (size_t)S * 64;
  float* res     = ws; ws += (size_t)S * 64;
  float* aff     = ws; ws += E;
  float* segsum  = ws; ws += S;
  unsigned* segmax = (unsigned*)ws;

  k_init<<<dim3((S * 64 + 255) / 256), dim3(256), 0, stream>>>(
      emb_sum, cnt, segsum, segmax, res, S);

  k_point<<<dim3((Np + 16 * PK_WAVES - 1) / (16 * PK_WAVES)), dim3(32 * PK_WAVES),
            0, stream>>>(feats, imap, spof, HEADP(7), o_sem, emb_sum, cnt, Np);

  k_sp<<<dim3((S + 16 * SP_WAVES - 1) / (16 * SP_WAVES)), dim3(32 * SP_WAVES),
         0, stream>>>(emb_sum, cnt, W_ecc, F(47), F(48), F(49),
                      HEADP(13), HEADP(19), HEADP(25), HEADP(31),
                      ecc, q, k, v, o_spsem, o_spoff, o_spocc, o_spsize, S);

  k_edge_aff<<<dim3((E + 255) / 256), dim3(256), 0, stream>>>(
      eu, ev, xyz, F(43), F(44), F(45), F(46), q, k, aff, segmax, E);

  k_edge_exp<<<dim3((E + 255) / 256), dim3(256), 0, stream>>>(
      eu, aff, segmax, segsum, E);

  k_edge_soft<<<dim3((E + 7) / 8), dim3(256), 0, stream>>>(
      eu, ev, aff, segsum, v, o_soft, res, E);

  k_final<<<dim3((S + 16 * SP_WAVES - 1) / (16 * SP_WAVES)), dim3(32 * SP_WAVES),
            0, stream>>>(ecc, res, HEADP(37), o_spdisc, S);
}